// LSTM_Univariate_120259084490
// MI455X (gfx1250) — compile-verified
//
#include <hip/hip_runtime.h>
#include <cstdint>

#define FEATS  512
#define FB     128   // features per block
#define T_TILE 64    // timesteps staged per LDS tile
#define NWAVE  4     // 128 threads / wave32

// gfx1250 v_tanh_f32 (new CDNA5 transcendental). Prefer the builtin; fall back
// to inline asm if this clang doesn't declare it.
__device__ __forceinline__ float tanh_fast(float v) {
#if __has_builtin(__builtin_amdgcn_tanhf)
  return __builtin_amdgcn_tanhf(v);
#else
  float r;
  asm("v_tanh_f32 %0, %1" : "=v"(r) : "v"(v));
  return r;
#endif
}

__global__ __launch_bounds__(FB, 1)
void lstm_univariate_scan(const float* __restrict__ x,
                          const float* __restrict__ W_ih,
                          const float* __restrict__ W_hh,
                          const float* __restrict__ b_ih,
                          const float* __restrict__ b_hh,
                          const float* __restrict__ h0,
                          const float* __restrict__ c0,
                          float* __restrict__ out,
                          int ntiles)
{
  // double-buffered x tile: [parity][t][feat], row stride FB floats (512 B)
  __shared__ __align__(16) float xs[2][T_TILE * FB];

  const int tid  = threadIdx.x;
  const int lane = tid & 31;
  const int wv   = tid >> 5;                 // wave id within block (0..3)
  const int f    = blockIdx.x * FB + tid;    // global feature

  // --- per-feature parameters, gate order i,f,g,o -------------------------
  // sigmoid gates (i,f,o): sigma(z) = 0.5*tanh(z/2)+0.5 -> prescale by 0.5
  float wi_i = W_ih[f * 4 + 0] * 0.5f;
  float wi_f = W_ih[f * 4 + 1] * 0.5f;
  float wi_g = W_ih[f * 4 + 2];
  float wi_o = W_ih[f * 4 + 3] * 0.5f;
  float wh_i = W_hh[f * 4 + 0] * 0.5f;
  float wh_f = W_hh[f * 4 + 1] * 0.5f;
  float wh_g = W_hh[f * 4 + 2];
  float wh_o = W_hh[f * 4 + 3] * 0.5f;
  float b_i  = (b_ih[f * 4 + 0] + b_hh[f * 4 + 0]) * 0.5f;
  float b_f  = (b_ih[f * 4 + 1] + b_hh[f * 4 + 1]) * 0.5f;
  float b_g  = (b_ih[f * 4 + 2] + b_hh[f * 4 + 2]);
  float b_o  = (b_ih[f * 4 + 3] + b_hh[f * 4 + 3]) * 0.5f;

  float h = h0[f];
  float c = c0[f];

  // LDS byte offset of the staging buffer (low 32 bits of generic pointer)
  const uint32_t lds_base   = (uint32_t)(uintptr_t)(&xs[0][0]);
  const uint32_t gblock_off = (uint32_t)(blockIdx.x * FB) * 4u;

  // Issue one tile's async copies: each wave copies rows wv, wv+4, ...
  // One global_load_async_to_lds_b128 moves 32 lanes x 16 B = 512 B = 1 row.
  auto issue_tile = [&](int tile, int parity) {
    uint32_t lds = lds_base + (uint32_t)parity * (T_TILE * FB * 4u)
                 + (uint32_t)wv * (FB * 4u) + (uint32_t)lane * 16u;
    uint32_t gof = gblock_off
                 + (uint32_t)(tile * T_TILE + wv) * (FEATS * 4u)
                 + (uint32_t)lane * 16u;
#pragma unroll
    for (int r = 0; r < T_TILE / NWAVE; ++r) {   // 16 async instrs / wave
      asm volatile("global_load_async_to_lds_b128 %0, %1, %2"
                   :: "v"(lds), "v"(gof), "s"(x)
                   : "memory");
      lds += NWAVE * FB * 4u;
      gof += NWAVE * FEATS * 4u;
    }
  };

  issue_tile(0, 0);

  for (int tile = 0; tile < ntiles; ++tile) {
    const int par = tile & 1;
    if (tile + 1 < ntiles) {
      issue_tile(tile + 1, par ^ 1);
      // 16 outstanding = the next tile; current tile's 16 have completed
      asm volatile("s_wait_asynccnt 16" ::: "memory");
    } else {
      asm volatile("s_wait_asynccnt 0" ::: "memory");
    }
    __syncthreads();   // current tile visible to all waves

    const float* xb = &xs[par][0];
    float* op = out + (size_t)tile * T_TILE * FEATS + f;

    float xc = xb[tid];                       // x for step s (prefetched)
#pragma unroll 4
    for (int s = 0; s < T_TILE; ++s) {
      // prefetch next step's x from LDS (off the recurrence chain)
      float xn = (s + 1 < T_TILE) ? xb[(s + 1) * FB + tid] : 0.0f;

      // x-dependent gate halves (independent of h -> off-chain)
      float ai = fmaf(xc, wi_i, b_i);
      float af = fmaf(xc, wi_f, b_f);
      float ag = fmaf(xc, wi_g, b_g);
      float ao = fmaf(xc, wi_o, b_o);

      // recurrence-dependent part
      float zi = fmaf(h, wh_i, ai);
      float zf = fmaf(h, wh_f, af);
      float zg = fmaf(h, wh_g, ag);
      float zo = fmaf(h, wh_o, ao);

      float gi = fmaf(tanh_fast(zi), 0.5f, 0.5f);   // sigmoid(i)
      float gf = fmaf(tanh_fast(zf), 0.5f, 0.5f);   // sigmoid(f)
      float gg = tanh_fast(zg);                      // tanh(g)
      float go = fmaf(tanh_fast(zo), 0.5f, 0.5f);   // sigmoid(o)

      c = fmaf(gf, c, gi * gg);
      h = go * tanh_fast(c);

      op[s * FEATS] = h + h;                  // module returns 2*h (off-chain)
      xc = xn;
    }
    __syncthreads();   // all waves done reading this buffer before it is
                       // overwritten by the issue two iterations ahead
  }
}

extern "C" void kernel_launch(void* const* d_in, const int* in_sizes, int n_in,
                              void* d_out, int out_size, void* d_ws, size_t ws_size,
                              hipStream_t stream) {
  const float* x    = (const float*)d_in[0];
  const float* W_ih = (const float*)d_in[1];
  const float* W_hh = (const float*)d_in[2];
  const float* b_ih = (const float*)d_in[3];
  const float* b_hh = (const float*)d_in[4];
  const float* h0   = (const float*)d_in[5];
  const float* c0   = (const float*)d_in[6];
  float* out = (float*)d_out;

  const int seq    = in_sizes[0] / FEATS;   // 32768
  const int ntiles = seq / T_TILE;          // 512

  lstm_univariate_scan<<<FEATS / FB, FB, 0, stream>>>(
      x, W_ih, W_hh, b_ih, b_hh, h0, c0, out, ntiles);
}